// SNNModel_63075889709425
// MI455X (gfx1250) — compile-verified
//
#include <hip/hip_runtime.h>

typedef __attribute__((ext_vector_type(2))) float v2f;
typedef __attribute__((ext_vector_type(8))) float v8f;

#define T_STEPS 128
#define BATCH   2048
#define DIN     256
#define HID     1024
#define DOUT    2

#define BT 64            // batch rows per workgroup
#define HT 128           // hidden cols per workgroup
#define LDS_STRIDE 260   // 256 + 4 pad floats -> conflict-free b64 reads

__global__ __launch_bounds__(256, 1)
void snn_fused(const float* __restrict__ spikes,
               const float* __restrict__ W1,
               const float* __restrict__ b1,
               const float* __restrict__ W2,
               float* __restrict__ out)
{
    __shared__ float w1s[HT * LDS_STRIDE];

    const int tid  = threadIdx.x;
    const int wave = tid >> 5;
    const int lane = tid & 31;
    const int lm   = lane & 15;   // lane within half-wave
    const int lg   = lane >> 4;   // half-wave group (splits K for A/B, splits M for C)

    const int h0 = blockIdx.x * HT;   // hidden tile base (fast dim -> L2 reuse of spikes)
    const int b0 = blockIdx.y * BT;   // batch tile base

    // 8 waves arranged 4 (batch) x 2 (hidden); each wave: 16 rows x 64 cols = 4 C-subtiles
    const int waveM = wave & 3;
    const int waveN = wave >> 1 >> 1;  // wave >> 2
    const int bRow0 = b0 + waveM * 16;
    const int hW    = waveN * 64;      // local hidden base for this wave

    // ---- stage W1 tile (HT x DIN) into padded LDS once; reused all 128 steps ----
    for (int idx = tid; idx < HT * (DIN / 4); idx += 256) {
        const int row = idx >> 6;
        const int c4  = (idx & 63) << 2;
        const float4 v = *(const float4*)(W1 + (size_t)(h0 + row) * DIN + c4);
        float* dst = &w1s[row * LDS_STRIDE + c4];
        dst[0] = v.x; dst[1] = v.y; dst[2] = v.z; dst[3] = v.w;
    }
    __syncthreads();

    // per-lane constants: bias and fc2 weights for the 4 hidden columns this lane owns
    float b1v[4], w20v[4], w21v[4];
    const float* bB[4];    // LDS read base for B fragments, per subtile
    #pragma unroll
    for (int s = 0; s < 4; ++s) {
        const int h = h0 + hW + s * 16 + lm;
        b1v[s]  = b1[h];
        w20v[s] = W2[h];          // W2[0][h]
        w21v[s] = W2[HID + h];    // W2[1][h]
        bB[s]   = &w1s[(hW + s * 16 + lm) * LDS_STRIDE + 2 * lg];
    }

    // A-fragment base (spikes): row = bRow0+lm, K sub-offset 2*lg
    const size_t aOff   = (size_t)(bRow0 + lm) * DIN + 2 * lg;
    const size_t rowOff = (size_t)(bRow0 + lm) * DIN;   // for prefetch (full row)

    // LIF state, resident in VGPRs for all 128 steps: element e = s*8 + r
    float vst[32], ist[32], zs[32];
    #pragma unroll
    for (int e = 0; e < 32; ++e) { vst[e] = 0.f; ist[e] = 0.f; zs[e] = 0.f; }

    #pragma unroll 1
    for (int t = 0; t < T_STEPS; ++t) {
        const float* aPtr = spikes + (size_t)t * (BATCH * DIN) + aOff;

        // prefetch next step's spike rows (global_prefetch_b8); 4x128B per lane
        if (t + 1 < T_STEPS) {
            const char* pn = (const char*)(spikes + (size_t)(t + 1) * (BATCH * DIN) + rowOff);
            #pragma unroll
            for (int j = 0; j < 4; ++j)
                __builtin_prefetch(pn + (size_t)(lg * 4 + j) * 128, 0, 0);
        }

        // accumulators initialized with bias (broadcast across the 8 M-rows per lane)
        v8f acc[4];
        #pragma unroll
        for (int s = 0; s < 4; ++s)
            #pragma unroll
            for (int r = 0; r < 8; ++r)
                acc[s][r] = b1v[s];

        // x_tile += s_t[16x256] @ W1^T[256x64]  via 64 chained K=4 f32 WMMAs per subtile
        #pragma unroll 8
        for (int k0 = 0; k0 < DIN; k0 += 4) {
            const v2f a = *(const v2f*)(aPtr + k0);
            #pragma unroll
            for (int s = 0; s < 4; ++s) {
                const v2f b = *(const v2f*)(bB[s] + k0);
                acc[s] = __builtin_amdgcn_wmma_f32_16x16x4_f32(
                    false, a, false, b, (short)0, acc[s], false, false);
            }
        }

        // LIF dynamics (element-wise, fp32, exactly the reference recurrence)
        #pragma unroll
        for (int s = 0; s < 4; ++s) {
            #pragma unroll
            for (int r = 0; r < 8; ++r) {
                const int e = s * 8 + r;
                float x  = acc[s][r];
                x = x > 0.f ? x : 0.f;                          // relu
                const float vd = vst[e] + 0.1f * (ist[e] - vst[e]);  // v + dt*tau_mem_inv*((0-v)+i)
                const float id = ist[e] - 0.2f * ist[e];             // i - dt*tau_syn_inv*i
                const bool  sp = vd > 1.0f;                          // superspike fwd: (vd - 1 > 0)
                vst[e] = sp ? 0.f : vd;                              // reset to 0 on spike
                ist[e] = id + x;                                     // current injection
                zs[e] += sp ? 1.f : 0.f;
            }
        }
    }

    // ---- out[b][d] += (zsum/T) * W2[d][h], reduced over this tile's hidden cols ----
    #pragma unroll
    for (int r = 0; r < 8; ++r) {
        float p0 = 0.f, p1 = 0.f;
        #pragma unroll
        for (int s = 0; s < 4; ++s) {
            p0 += zs[s * 8 + r] * w20v[s];
            p1 += zs[s * 8 + r] * w21v[s];
        }
        // butterfly reduce across the 16 lanes of each half-wave (same batch row)
        #pragma unroll
        for (int off = 1; off < 16; off <<= 1) {
            p0 += __shfl_xor(p0, off, 32);
            p1 += __shfl_xor(p1, off, 32);
        }
        if (lm == 0) {
            const int b = bRow0 + r + 8 * lg;   // C layout: VGPR r -> M=r (lg=0) / M=r+8 (lg=1)
            atomicAdd(out + b * 2 + 0, p0 * (1.0f / 128.0f));
            atomicAdd(out + b * 2 + 1, p1 * (1.0f / 128.0f));
        }
    }
}

// out is poisoned by the harness: initialize to b2 before the atomic accumulation.
__global__ void snn_init_out(const float* __restrict__ b2, float* __restrict__ out)
{
    const int idx = blockIdx.x * 256 + threadIdx.x;
    if (idx < BATCH * DOUT) out[idx] = b2[idx & 1];
}

extern "C" void kernel_launch(void* const* d_in, const int* in_sizes, int n_in,
                              void* d_out, int out_size, void* d_ws, size_t ws_size,
                              hipStream_t stream)
{
    const float* spikes = (const float*)d_in[0];  // [128, 2048, 256]
    const float* W1     = (const float*)d_in[1];  // [1024, 256]
    const float* b1     = (const float*)d_in[2];  // [1024]
    const float* W2     = (const float*)d_in[3];  // [2, 1024]
    const float* b2     = (const float*)d_in[4];  // [2]
    float* out = (float*)d_out;                   // [2048, 2]

    snn_init_out<<<(BATCH * DOUT + 255) / 256, 256, 0, stream>>>(b2, out);

    dim3 grid(HID / HT, BATCH / BT);  // (8, 32): H-tiles fastest -> spike tiles reused in L2
    snn_fused<<<grid, 256, 0, stream>>>(spikes, W1, b1, W2, out);
}